// GNNFiLM_16544214024609
// MI455X (gfx1250) — compile-verified
//
#include <hip/hip_runtime.h>

typedef float v2f __attribute__((ext_vector_type(2)));
typedef float v8f __attribute__((ext_vector_type(8)));

#define IN_FT  128
#define OUT_FT 128

// ---------------------------------------------------------------------------
// Kernel 1: zero the aggregation buffer (d_out used in-place as accumulator)
// ---------------------------------------------------------------------------
__global__ void gnnfilm_zero(float* __restrict__ p, int n4) {
  int i = blockIdx.x * blockDim.x + threadIdx.x;
  if (i < n4) {
    ((float4*)p)[i] = make_float4(0.f, 0.f, 0.f, 0.f);
  }
}

// ---------------------------------------------------------------------------
// Kernel 2: seq_fts = seq @ W_fc  via V_WMMA_F32_16X16X4_F32
//   one wave -> one 16x16 C tile; 8 waves/block -> 16-row x 128-col stripe
//   N = 50000 = 3125 * 16 exact, so every wave is full (EXEC all-1s).
// A frag (16x4):  lane l holds row m0+(l&15), K = kc*4 + (l>>4)*2 + {0,1}
// B frag (4x16):  vgpr j, lane l holds W[kc*4 + (l>>4)*2 + j][n0 + (l&15)]
// C tile (16x16): vgpr v, lane l -> row m0 + v + (l>>4)*8, col n0 + (l&15)
// ---------------------------------------------------------------------------
__global__ __launch_bounds__(256) void gnnfilm_gemm_wmma(
    const float* __restrict__ seq, const float* __restrict__ W,
    float* __restrict__ seq_fts) {
  const int lane = threadIdx.x & 31;
  const int wave = threadIdx.x >> 5;
  const int m0   = blockIdx.x << 4;
  const int n0   = wave << 4;
  const int half = lane >> 4;   // 0 or 1
  const int l15  = lane & 15;

  const float* __restrict__ arow = seq + (size_t)(m0 + l15) * IN_FT;
  const float* __restrict__ bcol = W + n0 + l15;

  v8f c = {0.f, 0.f, 0.f, 0.f, 0.f, 0.f, 0.f, 0.f};

#pragma unroll
  for (int kc = 0; kc < IN_FT / 4; ++kc) {
    const int kb = kc * 4 + half * 2;
    v2f a = *(const v2f*)(arow + kb);           // contiguous 8B load
    v2f b;
    b.x = bcol[(size_t)(kb + 0) * OUT_FT];
    b.y = bcol[(size_t)(kb + 1) * OUT_FT];
    c = __builtin_amdgcn_wmma_f32_16x16x4_f32(
        /*neg_a=*/false, a, /*neg_b=*/false, b,
        /*c_mod=*/(short)0, c, /*reuse_a=*/false, /*reuse_b=*/false);
  }

  float* __restrict__ crow =
      seq_fts + (size_t)(m0 + half * 8) * OUT_FT + n0 + l15;
#pragma unroll
  for (int v = 0; v < 8; ++v)
    crow[(size_t)v * OUT_FT] = c[v];
}

// ---------------------------------------------------------------------------
// Kernel 3: SpMM scatter. One wave per edge: 32 lanes x float4 = 128 feats.
//   Coalesced 512B gather from seq_fts[col] (L2-resident: 25.6MB << 192MB L2)
//   then 4 contiguous global_atomic_add_f32 into agg[row].
// ---------------------------------------------------------------------------
__global__ __launch_bounds__(256) void gnnfilm_spmm(
    const float* __restrict__ seq_fts, const float* __restrict__ ew,
    const int* __restrict__ erow, const int* __restrict__ ecol,
    float* __restrict__ agg, int E) {
  int tid = blockIdx.x * blockDim.x + threadIdx.x;
  int e = tid >> 5;
  if (e >= E) return;
  int f4 = (tid & 31) << 2;

  float w = ew[e];
  int cs  = ecol[e];
  int rd  = erow[e];

  float4 v = *(const float4*)(seq_fts + (size_t)cs * OUT_FT + f4);
  float* dst = agg + (size_t)rd * OUT_FT + f4;
  atomicAdd(dst + 0, w * v.x);
  atomicAdd(dst + 1, w * v.y);
  atomicAdd(dst + 2, w * v.z);
  atomicAdd(dst + 3, w * v.w);
}

// ---------------------------------------------------------------------------
// Kernel 4: FiLM + bias + residual + PReLU, in place on d_out.
//   gamma = W_gamma[type] + b_gamma ; beta = W_beta[type] + b_beta
//   out = prelu(gamma*agg + beta + bias + seq_fts)
// ---------------------------------------------------------------------------
__global__ __launch_bounds__(256) void gnnfilm_finalize(
    const float* __restrict__ seq_fts,
    const float* __restrict__ Wg, const float* __restrict__ bg,
    const float* __restrict__ Wb, const float* __restrict__ bb,
    const float* __restrict__ bias, const float* __restrict__ prelu_w,
    const int* __restrict__ node_type, float* __restrict__ out, int N) {
  int tid = blockIdx.x * blockDim.x + threadIdx.x;
  int n = tid >> 5;
  if (n >= N) return;
  int f4 = (tid & 31) << 2;

  int t = node_type[n];
  float slope = prelu_w[0];

  float4 g  = *(const float4*)(Wg + (size_t)t * OUT_FT + f4);
  float4 b  = *(const float4*)(Wb + (size_t)t * OUT_FT + f4);
  float4 gb = *(const float4*)(bg + f4);
  float4 bbv= *(const float4*)(bb + f4);
  float4 bi = *(const float4*)(bias + f4);
  float4 sf = *(const float4*)(seq_fts + (size_t)n * OUT_FT + f4);
  float4 ag = *(const float4*)(out + (size_t)n * OUT_FT + f4);

  float o0 = (g.x + gb.x) * ag.x + (b.x + bbv.x) + bi.x + sf.x;
  float o1 = (g.y + gb.y) * ag.y + (b.y + bbv.y) + bi.y + sf.y;
  float o2 = (g.z + gb.z) * ag.z + (b.z + bbv.z) + bi.z + sf.z;
  float o3 = (g.w + gb.w) * ag.w + (b.w + bbv.w) + bi.w + sf.w;

  float4 r;
  r.x = o0 >= 0.f ? o0 : slope * o0;
  r.y = o1 >= 0.f ? o1 : slope * o1;
  r.z = o2 >= 0.f ? o2 : slope * o2;
  r.w = o3 >= 0.f ? o3 : slope * o3;
  *(float4*)(out + (size_t)n * OUT_FT + f4) = r;
}

// ---------------------------------------------------------------------------
extern "C" void kernel_launch(void* const* d_in, const int* in_sizes, int n_in,
                              void* d_out, int out_size, void* d_ws, size_t ws_size,
                              hipStream_t stream) {
  const float* seq      = (const float*)d_in[0];
  const float* ew       = (const float*)d_in[1];
  const float* W_fc     = (const float*)d_in[2];
  const float* W_gamma  = (const float*)d_in[3];
  const float* b_gamma  = (const float*)d_in[4];
  const float* W_beta   = (const float*)d_in[5];
  const float* b_beta   = (const float*)d_in[6];
  const float* bias     = (const float*)d_in[7];
  const float* prelu_w  = (const float*)d_in[8];
  const int*   edge_row = (const int*)d_in[9];
  const int*   edge_col = (const int*)d_in[10];
  const int*   node_typ = (const int*)d_in[11];

  const int N = in_sizes[0] / IN_FT;   // 50000
  const int E = in_sizes[1];           // 800000

  float* out     = (float*)d_out;          // agg accumulator, finalized in place
  float* seq_fts = (float*)d_ws;           // N*128 floats = 25.6 MB scratch

  // 1) zero accumulator
  {
    int n4 = N * OUT_FT / 4;
    gnnfilm_zero<<<(n4 + 255) / 256, 256, 0, stream>>>(out, n4);
  }
  // 2) GEMM: one block per 16 rows (N divisible by 16)
  gnnfilm_gemm_wmma<<<N / 16, 256, 0, stream>>>(seq, W_fc, seq_fts);
  // 3) SpMM scatter: one wave per edge
  {
    long long threads = (long long)E * 32;
    int blocks = (int)((threads + 255) / 256);
    gnnfilm_spmm<<<blocks, 256, 0, stream>>>(seq_fts, ew, edge_row, edge_col, out, E);
  }
  // 4) FiLM + PReLU finalize: one wave per node
  {
    long long threads = (long long)N * 32;
    int blocks = (int)((threads + 255) / 256);
    gnnfilm_finalize<<<blocks, 256, 0, stream>>>(seq_fts, W_gamma, b_gamma,
                                                 W_beta, b_beta, bias, prelu_w,
                                                 node_typ, out, N);
  }
}